// QuantumLayer_36610301231485
// MI455X (gfx1250) — compile-verified
//
#include <hip/hip_runtime.h>

// CDNA5 / gfx1250. wave32. WMMA f32 16x16x4.
typedef __attribute__((ext_vector_type(2))) float v2f;
typedef __attribute__((ext_vector_type(8))) float v8f;

#define NQ    10
#define NL    2
#define DIM   1024      // 2^NQ amplitudes
#define BATCH 4096

// ---------------------------------------------------------------------------
// Kernel 1: build circuit unitary U by simulating each basis state |k>.
// Writes U^T (B matrix for the GEMM) re/im in K-pair swizzled layout:
//   element (k, n)  ->  ((k>>1)*DIM + n)*2 + (k&1)
// so a v2f load fetches B[kk][n], B[kk+1][n] contiguously.
// ---------------------------------------------------------------------------
__global__ __launch_bounds__(128) void k_build_unitary(
    const float* __restrict__ weights, float* __restrict__ Ur, float* __restrict__ Ui) {
  __shared__ float re[DIM];
  __shared__ float im[DIM];
  const int k   = blockIdx.x;
  const int tid = threadIdx.x;

  for (int j = tid; j < DIM; j += 128) { re[j] = (j == k) ? 1.f : 0.f; im[j] = 0.f; }
  __syncthreads();

  for (int l = 0; l < NL; ++l) {
    // RX(theta) on every wire: [[c, -i s],[-i s, c]], theta = w*0.5
    for (int w = 0; w < NQ; ++w) {
      const float th = weights[l * NQ + w] * 0.5f;
      const float c = cosf(th);
      const float t = sinf(th);
      const int bp = 9 - w;                 // qubit w lives at bit (9-w)
      for (int p = tid; p < DIM / 2; p += 128) {
        const int j0 = ((p >> bp) << (bp + 1)) | (p & ((1 << bp) - 1));
        const int j1 = j0 | (1 << bp);
        const float r0 = re[j0], i0 = im[j0];
        const float r1 = re[j1], i1 = im[j1];
        re[j0] = c * r0 + t * i1;  im[j0] = c * i0 - t * r1;
        re[j1] = t * i0 + c * r1;  im[j1] = -t * r0 + c * i1;
      }
      __syncthreads();
    }
    // CNOT ring (0,1)(1,2)...(9,0): swap amps with control=1, target=0 <-> 1
    for (int g = 0; g < NQ; ++g) {
      const int bc = 9 - g;
      const int bt = 9 - ((g + 1) % NQ);
      const int lo = bc < bt ? bc : bt;
      const int hi = bc < bt ? bt : bc;
      for (int p = tid; p < DIM / 4; p += 128) {
        int j = ((p >> lo) << (lo + 1)) | (p & ((1 << lo) - 1));
        j = ((j >> hi) << (hi + 1)) | (j & ((1 << hi) - 1));
        j |= (1 << bc);                     // control bit = 1, target bit = 0
        const int j2 = j | (1 << bt);
        const float r = re[j], i = im[j];
        re[j] = re[j2]; im[j] = im[j2];
        re[j2] = r;     im[j2] = i;
      }
      __syncthreads();
    }
  }
  for (int n = tid; n < DIM; n += 128) {
    const size_t off = ((size_t)(k >> 1) * DIM + (size_t)n) * 2 + (size_t)(k & 1);
    Ur[off] = re[n];
    Ui[off] = im[n];
  }
}

// ---------------------------------------------------------------------------
// Kernel 2: initial product states. S[b][j] = prod_w (bit_w(j) ? sin : cos)(x/2)
// qubit w <-> bit (9-w)  (qubit 0 most significant, C-order flatten).
// ---------------------------------------------------------------------------
__global__ __launch_bounds__(256) void k_build_states(
    const float* __restrict__ x, float* __restrict__ S) {
  const int b   = blockIdx.x;
  const int tid = threadIdx.x;
  float c[NQ], s[NQ];
#pragma unroll
  for (int i = 0; i < NQ; ++i) {
    const float h = x[b * NQ + i] * 0.5f;
    c[i] = cosf(h);
    s[i] = sinf(h);
  }
  for (int j = tid; j < DIM; j += 256) {
    float prod = 1.f;
#pragma unroll
    for (int i = 0; i < NQ; ++i) prod *= ((j >> (9 - i)) & 1) ? s[i] : c[i];
    S[(size_t)b * DIM + j] = prod;
  }
}

// ---------------------------------------------------------------------------
// Kernel 3: C = S * U^T via WMMA f32 16x16x4, fused |C|^2 and Z reduction.
// 256 threads = 8 waves; each wave owns a 16(M) x 16(N) tile, 8 N-tiles each.
// Deterministic: no atomics, fixed-order reductions.
// ---------------------------------------------------------------------------
__global__ __launch_bounds__(256) void k_gemm_z(
    const float* __restrict__ S, const float* __restrict__ Ur,
    const float* __restrict__ Ui, float* __restrict__ out) {
  __shared__ float patch[8][16][16];   // per-wave |amp|^2 tile
  __shared__ float zpart[8][16][10];   // per-wave per-row z partials

  const int tid  = threadIdx.x;
  const int wv   = tid >> 5;
  const int lane = tid & 31;
  const int half = lane >> 4;          // 0: lanes 0-15, 1: lanes 16-31
  const int m    = lane & 15;          // A-row / B-col / patch index
  const int b0   = blockIdx.x * 16;

  // A fragment (16x4 f32): lane holds A[m][k+2*half], A[m][k+2*half+1]
  const float* Arow = S + (size_t)(b0 + m) * DIM + (size_t)(2 * half);

  float zacc[10];
#pragma unroll
  for (int i = 0; i < 10; ++i) zacc[i] = 0.f;

  for (int t = 0; t < 8; ++t) {
    const int n0 = (wv * 8 + t) * 16;
    const int n  = n0 + m;
    // B fragment (4x16 f32): lane holds B[k+2*half][n], B[k+2*half+1][n]
    const float* Brow  = Ur + ((size_t)(half * DIM + n)) * 2;
    const float* Birow = Ui + ((size_t)(half * DIM + n)) * 2;

    v8f cre = {};
    v8f cim = {};
    for (int k = 0; k < DIM; k += 4) {
      const v2f a  = *(const v2f*)(Arow + k);
      const v2f br = *(const v2f*)(Brow + (size_t)(k >> 1) * (DIM * 2));
      const v2f bi = *(const v2f*)(Birow + (size_t)(k >> 1) * (DIM * 2));
      cre = __builtin_amdgcn_wmma_f32_16x16x4_f32(false, a, false, br, (short)0,
                                                  cre, false, false);
      cim = __builtin_amdgcn_wmma_f32_16x16x4_f32(false, a, false, bi, (short)0,
                                                  cim, false, false);
    }
    // C/D layout: VGPR v, lane l -> row v + 8*(l/16), col l%16
#pragma unroll
    for (int v = 0; v < 8; ++v) {
      const float p = cre[v] * cre[v] + cim[v] * cim[v];
      patch[wv][v + 8 * half][m] = p;
    }
    __syncthreads();
    if (half == 0) {  // lane m owns output row b0+m
#pragma unroll
      for (int n2 = 0; n2 < 16; ++n2) {
        const float p = patch[wv][m][n2];
        const int j = n0 + n2;
#pragma unroll
        for (int i = 0; i < 10; ++i) zacc[i] += ((j >> (9 - i)) & 1) ? -p : p;
      }
    }
    __syncthreads();
  }

  if (half == 0) {
#pragma unroll
    for (int i = 0; i < 10; ++i) zpart[wv][m][i] = zacc[i];
  }
  __syncthreads();
  if (tid < 160) {
    const int r = tid / 10, i = tid % 10;
    float sum = 0.f;
#pragma unroll
    for (int w = 0; w < 8; ++w) sum += zpart[w][r][i];
    out[(size_t)(b0 + r) * 10 + i] = sum;
  }
}

// ---------------------------------------------------------------------------
// Workspace layout: Ur 4MB | Ui 4MB | S 16MB  (24MB total, fits L2 entirely)
// ---------------------------------------------------------------------------
extern "C" void kernel_launch(void* const* d_in, const int* in_sizes, int n_in,
                              void* d_out, int out_size, void* d_ws, size_t ws_size,
                              hipStream_t stream) {
  const float* x       = (const float*)d_in[0];  // (4096, 10)
  const float* weights = (const float*)d_in[1];  // (2, 10)
  float* out = (float*)d_out;                    // (4096, 10)

  char* ws  = (char*)d_ws;
  float* Ur = (float*)ws;
  float* Ui = (float*)(ws + (size_t)DIM * DIM * sizeof(float));
  float* S  = (float*)(ws + (size_t)DIM * DIM * sizeof(float) * 2);

  k_build_unitary<<<DIM, 128, 0, stream>>>(weights, Ur, Ui);
  k_build_states<<<BATCH, 256, 0, stream>>>(x, S);
  k_gemm_z<<<BATCH / 16, 256, 0, stream>>>(S, Ur, Ui, out);
}